// SimpleGCN_5342939316803
// MI455X (gfx1250) — compile-verified
//
#include <hip/hip_runtime.h>

typedef __attribute__((ext_vector_type(2))) float v2f;
typedef __attribute__((ext_vector_type(8))) float v8f;

static __device__ __forceinline__ void atomAddF(float* p, float v) {
    unsafeAtomicAdd(p, v);   // hardware global_atomic_add_f32
}

// ---------------------------------------------------------------- utilities
__global__ void fill_f32(float* __restrict__ p, float v, long n) {
    long i = (long)blockIdx.x * blockDim.x + threadIdx.x;
    if (i < n) p[i] = v;
}

__global__ void deg_accum(const int* __restrict__ dst, float* __restrict__ deg, long E) {
    long e = (long)blockIdx.x * blockDim.x + threadIdx.x;
    if (e < E) atomAddF(&deg[dst[e]], 1.0f);
}

__global__ void to_rsqrt(float* __restrict__ p, long n) {
    long i = (long)blockIdx.x * blockDim.x + threadIdx.x;
    if (i < n) p[i] = rsqrtf(p[i]);
}

// ---------------------------------------------------------------- WMMA GEMM
// H[nrows x NCOLS] = X[nrows x K] @ W[K x NCOLS], fp32 exact via
// V_WMMA_F32_16X16X4_F32. One wave per 16-row tile; all NTILES column tiles
// accumulated simultaneously so X is read exactly once.
//
// B staging: W is packed in LDS as v2f pairs sW[kp][n] = {W[2kp][n],
// W[2kp+1][n]} so each B fragment is a single naturally-aligned ds_load_b64
// landing directly in the even VGPR pair WMMA consumes (no repack movs).
// Row stride S is chosen with S % 32 == 16 (v2f units) so the two half-waves
// (rows kp vs kp+1, same column range) map to disjoint halves of the 64
// LDS banks -> conflict-free b64 reads.
template <int K, int NCOLS, int NTILES>
__global__ __launch_bounds__(256) void gemm_wmma_f32(
    const float* __restrict__ X, const float* __restrict__ W,
    float* __restrict__ H, long nrows) {
    constexpr int NT16 = NTILES * 16;
    constexpr int S    = (NT16 % 32 == 16) ? NT16 : NT16 + 16;  // bank-split pad
    __shared__ v2f sW[K / 2][S];

    const int tid = threadIdx.x;
    for (int idx = tid; idx < (K / 2) * NT16; idx += 256) {
        int kp = idx / NT16;
        int n  = idx % NT16;
        v2f val;
        val.x = (n < NCOLS) ? W[(long)(2 * kp) * NCOLS + n] : 0.0f;
        val.y = (n < NCOLS) ? W[(long)(2 * kp + 1) * NCOLS + n] : 0.0f;
        sW[kp][n] = val;
    }
    __syncthreads();

    const int wave   = tid >> 5;
    const int lane   = tid & 31;
    const long mtile = (long)blockIdx.x * 8 + wave;
    const long row0  = mtile * 16;
    if (row0 >= nrows) return;            // wave-uniform: EXEC stays all-1s

    const int mr = lane & 15;             // A row / B col / C col within tile
    const int kh = lane >> 4;             // K-pair sub-offset within group of 4
    const int hi = kh * 8;                // C/D row offset

    long arow = row0 + mr;
    if (arow >= nrows) arow = nrows - 1;  // per-lane clamp (loads only)
    const v2f* __restrict__ xrow = (const v2f*)(X + arow * K);

    v8f acc[NTILES];
#pragma unroll
    for (int t = 0; t < NTILES; ++t) acc[t] = (v8f)0.0f;

#pragma unroll
    for (int kp0 = 0; kp0 < K / 2; kp0 += 2) {
        v2f a = xrow[kp0 + kh];           // A: vgpr j holds K = 2*(kp0+kh)+j
#pragma unroll
        for (int t = 0; t < NTILES; ++t) {
            v2f b = sW[kp0 + kh][t * 16 + mr];   // single b64 -> even VGPR pair
            acc[t] = __builtin_amdgcn_wmma_f32_16x16x4_f32(
                false, a, false, b, (short)0, acc[t], false, false);
        }
    }

#pragma unroll
    for (int t = 0; t < NTILES; ++t) {
        int col = t * 16 + mr;
        if (col >= NCOLS) continue;
#pragma unroll
        for (int r = 0; r < 8; ++r) {     // C/D: vgpr r holds row r + hi
            long m = row0 + r + hi;
            if (m < nrows) H[m * NCOLS + col] = acc[t][r];
        }
    }
}

// ---------------------------------------------------------------- scatter
// thread = (edge, channel). Lanes cover contiguous channels -> coalesced
// gather of h[s*C..] and contiguous atomic targets (single L2 sector).
template <int C>
__global__ void scatter_add(const int* __restrict__ src, const int* __restrict__ dst,
                            const float* __restrict__ dinv, const float* __restrict__ h,
                            float* __restrict__ agg, long nwork) {
    long t = (long)blockIdx.x * blockDim.x + threadIdx.x;
    if (t >= nwork) return;
    long e = t / C;
    int  c = (int)(t - e * C);
    int  s = src[e], d = dst[e];
    float norm = dinv[s] * dinv[d];
    atomAddF(&agg[(long)d * C + c], h[(long)s * C + c] * norm);
}

// ---------------------------------------------------------------- epilogue
// out = [relu]( agg + h * dinv^2 + bias ), in place over agg.
template <int C, bool RELU>
__global__ void finalize(const float* __restrict__ h, const float* __restrict__ dinv,
                         const float* __restrict__ bias, float* __restrict__ out,
                         long nwork) {
    long t = (long)blockIdx.x * blockDim.x + threadIdx.x;
    if (t >= nwork) return;
    long i = t / C;
    int  c = (int)(t - i * C);
    float di = dinv[i];
    float v  = out[t] + h[t] * di * di + bias[c];
    out[t] = RELU ? fmaxf(v, 0.0f) : v;
}

// ---------------------------------------------------------------- driver
extern "C" void kernel_launch(void* const* d_in, const int* in_sizes, int n_in,
                              void* d_out, int out_size, void* d_ws, size_t ws_size,
                              hipStream_t stream) {
    constexpr int F_IN = 128, HID = 64, NCLS = 40;

    const float* x  = (const float*)d_in[0];
    const int*   ei = (const int*)d_in[1];
    const float* W1 = (const float*)d_in[2];
    const float* b1 = (const float*)d_in[3];
    const float* W2 = (const float*)d_in[4];
    const float* b2 = (const float*)d_in[5];

    const long N = in_sizes[0] / F_IN;
    const long E = in_sizes[1] / 2;
    const int* srcI = ei;
    const int* dstI = ei + E;

    // workspace layout (floats), 1024-element aligned slices
    float* ws   = (float*)d_ws;
    long   offN = ((N + 1023) / 1024) * 1024;
    float* dinv = ws;                       // deg -> dinv, N
    float* h1   = dinv + offN;              // N * 64   (later reused for h2)
    float* agg1 = h1 + (long)N * HID;       // N * 64   (later holds out1)
    float* h2   = h1;                       // N * 40 fits in h1 slice
    float* out  = (float*)d_out;            // agg2 accumulated directly here

    const int B = 256;
    auto blocks = [](long n, int b) { return (int)((n + b - 1) / b); };

    // 1) degree (incl. self-loop) -> dinv = deg^{-1/2}
    fill_f32<<<blocks(N, B), B, 0, stream>>>(dinv, 1.0f, N);
    deg_accum<<<blocks(E, B), B, 0, stream>>>(dstI, dinv, E);
    to_rsqrt<<<blocks(N, B), B, 0, stream>>>(dinv, N);

    // 2) h1 = x @ W1   (WMMA f32, 6250 row-tiles, 8 waves/block)
    {
        long mtiles = (N + 15) / 16;
        gemm_wmma_f32<F_IN, HID, HID / 16>
            <<<blocks(mtiles, 8), B, 0, stream>>>(x, W1, h1, N);
    }

    // 3) agg1 = scatter(h1);  out1 = relu(agg1 + h1*dinv^2 + b1)  (in place)
    fill_f32<<<blocks(N * HID, B), B, 0, stream>>>(agg1, 0.0f, N * HID);
    scatter_add<HID><<<blocks(E * HID, B), B, 0, stream>>>(srcI, dstI, dinv, h1, agg1, E * HID);
    finalize<HID, true><<<blocks(N * HID, B), B, 0, stream>>>(h1, dinv, b1, agg1, N * HID);

    // 4) h2 = out1 @ W2   (40 cols -> 3 padded tiles)
    {
        long mtiles = (N + 15) / 16;
        gemm_wmma_f32<HID, NCLS, 3>
            <<<blocks(mtiles, 8), B, 0, stream>>>(agg1, W2, h2, N);
    }

    // 5) d_out = scatter(h2) + h2*dinv^2 + b2
    fill_f32<<<blocks(N * NCLS, B), B, 0, stream>>>(out, 0.0f, N * NCLS);
    scatter_add<NCLS><<<blocks(E * NCLS, B), B, 0, stream>>>(srcI, dstI, dinv, h2, out, E * NCLS);
    finalize<NCLS, false><<<blocks(N * NCLS, B), B, 0, stream>>>(h2, dinv, b2, out, N * NCLS);
}